// KnowledgeFusion_33165737460139
// MI455X (gfx1250) — compile-verified
//
#include <hip/hip_runtime.h>
#include <hip/hip_bf16.h>

typedef __attribute__((ext_vector_type(2))) float v2f;
typedef __attribute__((ext_vector_type(8))) float v8f;

#define NB   8
#define H    32
#define W    32
#define HW   1024
#define D    256
#define NOBJ 15
#define NM   16          // 15 object masks + full-image mask
#define LDA  260         // padded LDS stride (260 % 64 == 4 -> conflict-free)

// ---------------------------------------------------------------------------
// Kernel 0: repack Wp row-major [256][256] into row-pair-interleaved float2:
//   Wq[k2][n] = (Wp[2k2][n], Wp[2k2+1][n])
// so each WMMA B-fragment per lane is a single contiguous global_load_b64.
// One-shot 512 KB of traffic; negligible.
// ---------------------------------------------------------------------------
__global__ __launch_bounds__(256) void kf_repack_wp(
    const float* __restrict__ Wp,   // [256,256]
    float2* __restrict__ Wq)        // [128,256] float2
{
    const int idx = blockIdx.x * 256 + threadIdx.x;  // 0..32767
    const int k2  = idx >> 8;                        // 0..127
    const int n   = idx & 255;
    float2 v;
    v.x = Wp[(2 * k2 + 0) * D + n];
    v.y = Wp[(2 * k2 + 1) * D + n];
    Wq[k2 * D + n] = v;
}

// ---------------------------------------------------------------------------
// Kernel 1: injq[b][n2][o] = float2( embs_ext[b][2n2]@We[:,o],
//                                    embs_ext[b][2n2+1]@We[:,o] )
//   embs_ext[n<15] = embs[b][n], embs_ext[15] = mean over the 15 objects.
// Pairs two mask-channel rows per block so each We load feeds two dots and
// the paired-B layout is produced directly (coalesced float2 stores).
// ---------------------------------------------------------------------------
__global__ __launch_bounds__(256) void kf_inj_kernel(
    const float* __restrict__ embs,   // [8,15,256]
    const float* __restrict__ We,     // [256,256]
    float2* __restrict__ injq)        // [8,8,256] float2
{
    __shared__ float lE0[D];
    __shared__ float lE1[D];
    const int b  = blockIdx.x >> 3;
    const int n2 = blockIdx.x & 7;
    const int t  = threadIdx.x;

    const float* eb = embs + (size_t)b * NOBJ * D;
    const int n0 = 2 * n2;
    const int n1 = 2 * n2 + 1;

    lE0[t] = eb[n0 * D + t];
    if (n1 < NOBJ) {
        lE1[t] = eb[n1 * D + t];
    } else {                                   // n1 == 15: mean embedding
        float s = 0.f;
        #pragma unroll
        for (int j = 0; j < NOBJ; ++j) s += eb[j * D + t];
        lE1[t] = s * (1.0f / 15.0f);
    }
    __syncthreads();

    float a0 = 0.f, a1 = 0.f;
    #pragma unroll 8
    for (int k = 0; k < D; ++k) {
        const float w = We[k * D + t];
        a0 += lE0[k] * w;
        a1 += lE1[k] * w;
    }
    float2 r; r.x = a0; r.y = a1;
    injq[((size_t)b * 8 + n2) * D + t] = r;
}

// ---------------------------------------------------------------------------
// Kernel 2: fused  out = (Mnorm @ inj) + (patches @ Wp)
//   One workgroup = one (batch, 16-row tile). 8 waves; wave w owns col tiles
//   {2w, 2w+1}. Mask term folds 1/S into the A matrix (row-dependent only),
//   so both GEMMs chain into the same v_wmma_f32_16x16x4_f32 accumulators.
//   B fragments come from the paired-float2 layouts -> global_load_b64.
// ---------------------------------------------------------------------------
__global__ __launch_bounds__(256) void kf_main_kernel(
    const float*  __restrict__ patches,   // [8,32,32,256] == [8,1024,256]
    const int*    __restrict__ locations, // [8,15,4]
    const float2* __restrict__ Wq,        // [128,256] paired Wp
    const float2* __restrict__ injq,      // [8,8,256] paired inj
    float* __restrict__ out)              // [8,1024,256]
{
    __shared__ float lA[16 * LDA];   // patches tile [16][256], padded
    __shared__ float lM[16][NM];     // raw masks  m[row][n]
    __shared__ float lMA[16][NM];    // normalized m[row][n] / S[row]

    const int wg   = blockIdx.x;       // 0..511
    const int b    = wg >> 6;
    const int rt   = wg & 63;
    const int row0 = rt * 16;          // base flattened spatial index

    const int t    = threadIdx.x;
    const int lane = t & 31;
    const int wave = t >> 5;           // 0..7

    // ---- stage patches tile into LDS (16 rows x 256 f32, float4 moves) ----
    const float* pbase = patches + ((size_t)b * HW + row0) * D;
    #pragma unroll
    for (int i = t; i < 16 * 64; i += 256) {       // 1024 float4 total
        const int r  = i >> 6;
        const int c4 = i & 63;
        const float4 v = ((const float4*)pbase)[r * 64 + c4];
        *((float4*)&lA[r * LDA + c4 * 4]) = v;
    }

    // ---- build masks: thread t -> (row = t/16, n = t%16) ----
    {
        const int row = t >> 4;
        const int n   = t & 15;
        const int p   = row0 + row;
        const int y   = p >> 5;
        const int x   = p & 31;
        float m;
        if (n == NOBJ) {
            m = 1.0f;                                   // full-image mask
        } else {
            const int* L = locations + ((size_t)b * NOBJ + n) * 4;
            const int l0 = L[0], l1 = L[1], l2 = L[2], l3 = L[3];
            const int y0 = l0 - (l0 & 1);               // floor to mult of 2
            const int x0 = l1 - (l1 & 1);
            const int y1 = l2 + (2 - (l2 & 1));         // ceil past, per ref
            const int x1 = l3 + (2 - (l3 & 1));
            m = (y >= y0 && y < y1 && x >= x0 && x < x1) ? 1.0f : 0.0f;
        }
        lM[row][n] = m;
    }
    __syncthreads();
    {
        const int row = t >> 4;
        const int n   = t & 15;
        float S = 0.f;
        #pragma unroll
        for (int k = 0; k < NM; ++k) S += lM[row][k];
        lMA[row][n] = lM[row][n] / S;                   // S >= 1 always
    }
    __syncthreads();

    // ---- per-wave WMMA: A layout (16x4 f32): lanes 0-15 K={0,1}, 16-31 K={2,3}
    const int mrow  = lane & 15;     // M index for A reads
    const int khalf = lane >> 4;     // which K pair this lane holds
    const int ncol  = lane & 15;     // N index for B reads
    const int c0    = wave * 32;          // first col tile base
    const int c1    = wave * 32 + 16;     // second col tile base

    v8f acc0 = {};
    v8f acc1 = {};

    // mask-term GEMM: D += Mnorm[16x16] @ inj[16x256-slice], K = 16
    const float2* injb = injq + (size_t)b * 8 * D;
    #pragma unroll
    for (int kk = 0; kk < NM; kk += 4) {
        const int k2 = (kk >> 1) + khalf;           // paired-row index
        const int k0 = kk + 2 * khalf;
        v2f a;  a.x = lMA[mrow][k0];  a.y = lMA[mrow][k0 + 1];
        const float2 f0 = injb[k2 * D + c0 + ncol];
        const float2 f1 = injb[k2 * D + c1 + ncol];
        v2f b0; b0.x = f0.x; b0.y = f0.y;
        v2f b1; b1.x = f1.x; b1.y = f1.y;
        acc0 = __builtin_amdgcn_wmma_f32_16x16x4_f32(false, a, false, b0,
                                                     (short)0, acc0, false, false);
        acc1 = __builtin_amdgcn_wmma_f32_16x16x4_f32(false, a, false, b1,
                                                     (short)0, acc1, false, false);
    }

    // projection GEMM: D += patches_tile[16x256] @ Wp[256x256-slice], K = 256
    #pragma unroll 4
    for (int kk = 0; kk < D; kk += 4) {
        const int k2 = (kk >> 1) + khalf;           // paired-row index
        const int k0 = kk + 2 * khalf;
        v2f a;  a.x = lA[mrow * LDA + k0];  a.y = lA[mrow * LDA + k0 + 1];
        const float2 f0 = Wq[k2 * D + c0 + ncol];   // global_load_b64
        const float2 f1 = Wq[k2 * D + c1 + ncol];   // global_load_b64
        v2f b0; b0.x = f0.x; b0.y = f0.y;
        v2f b1; b1.x = f1.x; b1.y = f1.y;
        acc0 = __builtin_amdgcn_wmma_f32_16x16x4_f32(false, a, false, b0,
                                                     (short)0, acc0, false, false);
        acc1 = __builtin_amdgcn_wmma_f32_16x16x4_f32(false, a, false, b1,
                                                     (short)0, acc1, false, false);
    }

    // ---- store: C/D layout: VGPR r -> M = r + 8*khalf, N = lane&15 ----
    float* ob = out + ((size_t)b * HW + row0) * D;
    #pragma unroll
    for (int r = 0; r < 8; ++r) {
        const int row = r + 8 * khalf;
        ob[row * D + c0 + ncol] = acc0[r];
        ob[row * D + c1 + ncol] = acc1[r];
    }
}

// ---------------------------------------------------------------------------
extern "C" void kernel_launch(void* const* d_in, const int* in_sizes, int n_in,
                              void* d_out, int out_size, void* d_ws, size_t ws_size,
                              hipStream_t stream) {
    const float* patches   = (const float*)d_in[0];   // [8,32,32,256]
    const float* embs      = (const float*)d_in[1];   // [8,15,256]
    const int*   locations = (const int*)d_in[2];     // [8,15,4]
    const float* Wp        = (const float*)d_in[3];   // [256,256]
    const float* We        = (const float*)d_in[4];   // [256,256]
    float*       out       = (float*)d_out;           // [8,1024,256]

    // workspace layout: [0, 256KB) Wq ; [256KB, 384KB) injq
    float2* Wq   = (float2*)d_ws;                           // 128*256 float2
    float2* injq = (float2*)((char*)d_ws + 128 * D * sizeof(float2));  // 8*8*256

    kf_repack_wp<<<128, 256, 0, stream>>>(Wp, Wq);
    kf_inj_kernel<<<NB * 8, 256, 0, stream>>>(embs, We, injq);
    kf_main_kernel<<<NB * (HW / 16), 256, 0, stream>>>(patches, locations, Wq, injq, out);
}